// DemandPredictionModel_43920335569133
// MI455X (gfx1250) — compile-verified
//
#include <hip/hip_runtime.h>

#define N_NODES 50000
#define N_EDGES 800000
#define CH 256

typedef __attribute__((ext_vector_type(2))) float v2f;
typedef __attribute__((ext_vector_type(8))) float v8f;

// ---------------------------------------------------------------- utility
__global__ void zero_kernel(float* __restrict__ p, int n) {
    int i = blockIdx.x * blockDim.x + threadIdx.x;
    if (i < n) p[i] = 0.0f;
}

__global__ void deg_count_kernel(const int* __restrict__ dst, float* __restrict__ deg) {
    int e = blockIdx.x * blockDim.x + threadIdx.x;   // grid sized exactly to E
    atomicAdd(&deg[dst[e]], 1.0f);
}

__global__ void dinv_kernel(const float* __restrict__ deg, float* __restrict__ dinv, int n) {
    int i = blockIdx.x * blockDim.x + threadIdx.x;
    if (i < n) dinv[i] = rsqrtf(deg[i] + 1.0f);      // +1 accounts for the self loop
}

// ---------------------------------------------------------------- WMMA GEMM
// C[M x 256] = A[M x 256] @ W[256 x 256], fp32 via V_WMMA_F32_16X16X4_F32.
#define TM 128
#define TN 64
#define TK 32
#define LDK 36   // pad; keeps 16B alignment for float4 LDS stores

__global__ __launch_bounds__(256) void gemm_kernel(const float* __restrict__ A,
                                                   const float* __restrict__ W,
                                                   float* __restrict__ C, int M) {
    __shared__ __align__(16) float As[TM * LDK];
    __shared__ __align__(16) float Bs[TN * LDK];   // stored transposed: [n][k]

    const int tid  = threadIdx.x;
    const int lane = tid & 31;
    const int wave = tid >> 5;
    const int wm   = wave & 3;     // 4 waves along M (32 rows each)
    const int wn   = wave >> 2;    // 2 waves along N (32 cols each)
    const int m0   = blockIdx.x * TM;
    const int n0   = blockIdx.y * TN;

    const int l15  = lane & 15;
    const int koff = (lane >> 4) << 1;   // lanes 16-31 hold K+2,K+3

    // global->LDS mapping
    const int ar  = tid >> 1;            // A row 0..127
    const int ac  = (tid & 1) * 16;      // A col half: 0 or 16
    const int bn  = tid & 63;            // B col 0..63
    const int bk0 = (tid >> 6) * 8;      // B k-chunk base

    v8f acc[2][2] = {};

    for (int k0 = 0; k0 < CH; k0 += TK) {
        __syncthreads();
        {   // A tile: 128 rows x 32 k (row-clamped for the M tail)
            int grow = m0 + ar; if (grow >= M) grow = M - 1;
            const float4* gp = (const float4*)&A[grow * CH + k0 + ac];
            float4* sp = (float4*)&As[ar * LDK + ac];
            sp[0] = gp[0]; sp[1] = gp[1]; sp[2] = gp[2]; sp[3] = gp[3];
        }
        // B tile transposed: Bs[n][kk] = W[k0+kk][n0+n]  (coalesced global reads)
        #pragma unroll
        for (int j = 0; j < 8; ++j) {
            int kk = bk0 + j;
            Bs[bn * LDK + kk] = W[(k0 + kk) * CH + n0 + bn];
        }
        __syncthreads();

        #pragma unroll
        for (int k = 0; k < TK; k += 4) {
            v2f a0 = *(const v2f*)&As[(wm * 32 +      l15) * LDK + k + koff];
            v2f a1 = *(const v2f*)&As[(wm * 32 + 16 + l15) * LDK + k + koff];
            v2f b0 = *(const v2f*)&Bs[(wn * 32 +      l15) * LDK + k + koff];
            v2f b1 = *(const v2f*)&Bs[(wn * 32 + 16 + l15) * LDK + k + koff];
            acc[0][0] = __builtin_amdgcn_wmma_f32_16x16x4_f32(false, a0, false, b0, (short)0, acc[0][0], false, false);
            acc[0][1] = __builtin_amdgcn_wmma_f32_16x16x4_f32(false, a0, false, b1, (short)0, acc[0][1], false, false);
            acc[1][0] = __builtin_amdgcn_wmma_f32_16x16x4_f32(false, a1, false, b0, (short)0, acc[1][0], false, false);
            acc[1][1] = __builtin_amdgcn_wmma_f32_16x16x4_f32(false, a1, false, b1, (short)0, acc[1][1], false, false);
        }
    }

    // C/D layout: VGPR r -> M = r + 8*(lane>=16), N = lane&15
    const int rbase = (lane >> 4) * 8;
    #pragma unroll
    for (int i = 0; i < 2; ++i) {
        #pragma unroll
        for (int j = 0; j < 2; ++j) {
            int col = n0 + wn * 32 + j * 16 + l15;
            #pragma unroll
            for (int r = 0; r < 8; ++r) {
                int row = m0 + wm * 32 + i * 16 + rbase + r;
                if (row < M) C[row * CH + col] = acc[i][j][r];
            }
        }
    }
}

// ---------------------------------------------------------------- aggregation
// agg[n,c] = bias[c] + dinv[n]^2 * xw[n,c]   (bias + self-loop term)
__global__ __launch_bounds__(256) void agg_init_kernel(const float* __restrict__ xw,
                                                       const float* __restrict__ dinv,
                                                       const float* __restrict__ bias,
                                                       float* __restrict__ agg) {
    int gid = blockIdx.x * blockDim.x + threadIdx.x;   // grid exact: N*64
    int n = gid >> 6;
    int c = (gid & 63) * 4;
    float di = dinv[n];
    float s = di * di;
    float4 v = *(const float4*)&xw[n * CH + c];
    float4 b = *(const float4*)&bias[c];
    float4 o;
    o.x = b.x + s * v.x; o.y = b.y + s * v.y;
    o.z = b.z + s * v.z; o.w = b.w + s * v.w;
    *(float4*)&agg[n * CH + c] = o;
}

// agg[dst,c] += dinv[src]*dinv[dst] * xin[src,c]   (64 lanes x float4 per edge)
__global__ __launch_bounds__(256) void edge_scatter_kernel(const int* __restrict__ src,
                                                           const int* __restrict__ dst,
                                                           const float* __restrict__ dinv,
                                                           const float* __restrict__ xin,
                                                           float* __restrict__ agg) {
    int e = blockIdx.x * 4 + (threadIdx.x >> 6);       // grid exact: E/4
    int c = (threadIdx.x & 63) * 4;
    int s = src[e], d = dst[e];
    float nrm = dinv[s] * dinv[d];
    float4 v = *(const float4*)&xin[s * CH + c];
    float* p = &agg[d * CH + c];
    atomicAdd(p + 0, nrm * v.x);
    atomicAdd(p + 1, nrm * v.y);
    atomicAdd(p + 2, nrm * v.z);
    atomicAdd(p + 3, nrm * v.w);
}

__global__ __launch_bounds__(256) void relu_kernel(const float* __restrict__ in,
                                                   float* __restrict__ out) {
    int gid = blockIdx.x * blockDim.x + threadIdx.x;   // grid exact: N*CH/4
    float4 v = *(const float4*)&in[gid * 4];
    v.x = fmaxf(v.x, 0.f); v.y = fmaxf(v.y, 0.f);
    v.z = fmaxf(v.z, 0.f); v.w = fmaxf(v.w, 0.f);
    *(float4*)&out[gid * 4] = v;
}

// ---------------------------------------------------------------- final head
// out[n] = sum_c agg2[n,c]*Wf[c] + sum_c relu(tf@Wt+bt)[c]*Wf[256+c] + bf
__global__ __launch_bounds__(256) void final_kernel(const float* __restrict__ agg2,
                                                    const float* __restrict__ tf,
                                                    const float* __restrict__ Wt,
                                                    const float* __restrict__ bt,
                                                    const float* __restrict__ Wf,
                                                    const float* __restrict__ bf,
                                                    float* __restrict__ out) {
    __shared__ float sWt[2 * CH];
    __shared__ float sbt[CH];
    __shared__ float sWf[2 * CH];
    int tid = threadIdx.x;
    for (int i = tid; i < 2 * CH; i += 256) { sWt[i] = Wt[i]; sWf[i] = Wf[i]; }
    for (int i = tid; i < CH; i += 256) sbt[i] = bt[i];
    __syncthreads();

    int lane = tid & 31;
    int n = blockIdx.x * 8 + (tid >> 5);               // grid exact: N/8
    float t0 = tf[n * 2 + 0];
    float t1 = tf[n * 2 + 1];
    float acc = 0.f;
    #pragma unroll
    for (int j = 0; j < 8; ++j) {
        int c = lane + 32 * j;
        acc += agg2[n * CH + c] * sWf[c];
        float xt = fmaxf(t0 * sWt[c] + t1 * sWt[CH + c] + sbt[c], 0.f);
        acc += xt * sWf[CH + c];
    }
    #pragma unroll
    for (int off = 16; off > 0; off >>= 1) acc += __shfl_xor(acc, off, 32);
    if (lane == 0) out[n] = acc + bf[0];
}

// ---------------------------------------------------------------- launcher
extern "C" void kernel_launch(void* const* d_in, const int* in_sizes, int n_in,
                              void* d_out, int out_size, void* d_ws, size_t ws_size,
                              hipStream_t stream) {
    const float* x  = (const float*)d_in[0];
    const int*   ei = (const int*)d_in[1];
    const float* tf = (const float*)d_in[2];
    const float* W1 = (const float*)d_in[3];
    const float* b1 = (const float*)d_in[4];
    const float* W2 = (const float*)d_in[5];
    const float* b2 = (const float*)d_in[6];
    const float* Wt = (const float*)d_in[7];
    const float* bt = (const float*)d_in[8];
    const float* Wf = (const float*)d_in[9];
    const float* bf = (const float*)d_in[10];
    float* out = (float*)d_out;

    float* ws   = (float*)d_ws;
    float* deg  = ws;                              // N
    float* dinv = ws + N_NODES;                    // N
    float* B0   = ws + 2 * N_NODES;                // N*CH
    float* B1   = B0 + (size_t)N_NODES * CH;       // N*CH

    const int* src = ei;
    const int* dst = ei + N_EDGES;

    // degree / normalization
    zero_kernel<<<(N_NODES + 255) / 256, 256, 0, stream>>>(deg, N_NODES);
    deg_count_kernel<<<N_EDGES / 256, 256, 0, stream>>>(dst, deg);
    dinv_kernel<<<(N_NODES + 255) / 256, 256, 0, stream>>>(deg, dinv, N_NODES);

    dim3 gg((N_NODES + TM - 1) / TM, CH / TN);

    // layer 1: B0 = x@W1 ; B1 = b1 + selfloop + scatter ; B0 = relu(B1)
    gemm_kernel<<<gg, 256, 0, stream>>>(x, W1, B0, N_NODES);
    agg_init_kernel<<<N_NODES * 64 / 256, 256, 0, stream>>>(B0, dinv, b1, B1);
    edge_scatter_kernel<<<N_EDGES / 4, 256, 0, stream>>>(src, dst, dinv, B0, B1);
    relu_kernel<<<(size_t)N_NODES * CH / 4 / 256, 256, 0, stream>>>(B1, B0);

    // layer 2: B1 = h@W2 ; B0 = b2 + selfloop + scatter
    gemm_kernel<<<gg, 256, 0, stream>>>(B0, W2, B1, N_NODES);
    agg_init_kernel<<<N_NODES * 64 / 256, 256, 0, stream>>>(B1, dinv, b2, B0);
    edge_scatter_kernel<<<N_EDGES / 4, 256, 0, stream>>>(src, dst, dinv, B1, B0);

    // head: time MLP + concat + 512->1 projection
    final_kernel<<<N_NODES / 8, 256, 0, stream>>>(B0, tf, Wt, bt, Wf, bf, out);
}